// GIN_20830591386316
// MI455X (gfx1250) — compile-verified
//
#include <hip/hip_runtime.h>
#include <math.h>

typedef __attribute__((ext_vector_type(2))) float v2f;
typedef __attribute__((ext_vector_type(8))) float v8f;

#define BN_EPS 1e-5f
#define NCLS 10

// ---------------- zero ----------------
__global__ void k_zero(float* __restrict__ p, long long n) {
  long long stride = (long long)gridDim.x * blockDim.x;
  for (long long i = (long long)blockIdx.x * blockDim.x + threadIdx.x; i < n; i += stride)
    p[i] = 0.0f;
}

// ---------------- edge scatter-add: agg[dst] += h[src] ----------------
// 16 threads per edge, each handles 4 features (float4 gather, 4 f32 atomics)
__global__ void k_agg(const float* __restrict__ h, const long long* __restrict__ ei,
                      float* agg, int E) {
  long long total = (long long)E * 16;
  long long stride = (long long)gridDim.x * blockDim.x;
  for (long long t = (long long)blockIdx.x * blockDim.x + threadIdx.x; t < total; t += stride) {
    long long e  = t >> 4;
    int       f4 = (int)(t & 15) * 4;
    long long s = ei[e];
    long long d = ei[(long long)E + e];
    const float4 v = *(const float4*)(h + s * 64 + f4);
    float* dp = agg + d * 64 + f4;
    atomicAdd(dp + 0, v.x);
    atomicAdd(dp + 1, v.y);
    atomicAdd(dp + 2, v.z);
    atomicAdd(dp + 3, v.w);
  }
}

// ---------------- WMMA GEMM: C[nrows x 64] = (A1 (+A2)) @ B[64x64] + bias ----------------
// One wave per 16-row strip. W staged in LDS in K-pair-interleaved layout:
//   Bp[(k>>1)*128 + n*2 + (k&1)]
// so each lane's {B[k0][n], B[k0+1][n]} pair is one aligned 8-byte ds_load_b64.
// 4 column tiles x 16 K-steps of v_wmma_f32_16x16x4_f32.
// In-place (Cout aliasing A1/A2) is safe: a wave writes only its own 16 rows
// and all of its A reads precede its stores.
#define GEMM_WPB 8
template <bool HAS_A2>
__global__ void k_gemm64(const float* A1, const float* A2,
                         const float* __restrict__ B, const float* __restrict__ bias,
                         float* Cout, int nrows) {
  __shared__ float Bp[64 * 64];
  __shared__ float bs[64];
  for (int i = threadIdx.x; i < 64 * 64; i += blockDim.x) {
    int k = i >> 6, n = i & 63;
    Bp[(k >> 1) * 128 + n * 2 + (k & 1)] = B[i];
  }
  if (threadIdx.x < 64) bs[threadIdx.x] = bias[threadIdx.x];
  __syncthreads();

  int wave = threadIdx.x >> 5;
  int lane = threadIdx.x & 31;
  int strip = blockIdx.x * GEMM_WPB + wave;
  if (strip * 16 >= nrows) return;  // uniform per wave -> EXEC all-1s where WMMA runs

  size_t row0  = (size_t)strip * 16;
  int    m     = lane & 15;   // A row within tile
  int    khalf = lane >> 4;   // 0: K0/K1  1: K2/K3
  int    nl    = lane & 15;   // B/C column within tile

  v8f c0 = {}, c1 = {}, c2 = {}, c3 = {};
  const v2f* a1p = (const v2f*)(A1 + (row0 + m) * 64) + khalf;
  const v2f* a2p = HAS_A2 ? ((const v2f*)(A2 + (row0 + m) * 64) + khalf) : (const v2f*)0;
  const v2f* bp  = (const v2f*)Bp;

#pragma unroll
  for (int kb = 0; kb < 64; kb += 4) {
    int kp = (kb >> 1) + khalf;  // K-pair row index for this lane
    v2f a = a1p[kb >> 1];
    if (HAS_A2) {
      v2f a2 = a2p[kb >> 1];
      a.x += a2.x;
      a.y += a2.y;
    }
    v2f b0 = bp[kp * 64 +  0 + nl];
    v2f b1 = bp[kp * 64 + 16 + nl];
    v2f b2 = bp[kp * 64 + 32 + nl];
    v2f b3 = bp[kp * 64 + 48 + nl];

    c0 = __builtin_amdgcn_wmma_f32_16x16x4_f32(false, a, false, b0, (short)0, c0, false, false);
    c1 = __builtin_amdgcn_wmma_f32_16x16x4_f32(false, a, false, b1, (short)0, c1, false, false);
    c2 = __builtin_amdgcn_wmma_f32_16x16x4_f32(false, a, false, b2, (short)0, c2, false, false);
    c3 = __builtin_amdgcn_wmma_f32_16x16x4_f32(false, a, false, b3, (short)0, c3, false, false);
  }

  // C layout: VGPR r -> row (r + 8*(lane/16)), col = lane%16 within tile
  int mh = khalf * 8;
#pragma unroll
  for (int r = 0; r < 8; ++r) {
    size_t rowi = (row0 + r + mh) * 64;
    Cout[rowi +  0 + nl] = c0[r] + bs[ 0 + nl];
    Cout[rowi + 16 + nl] = c1[r] + bs[16 + nl];
    Cout[rowi + 32 + nl] = c2[r] + bs[32 + nl];
    Cout[rowi + 48 + nl] = c3[r] + bs[48 + nl];
  }
}

// ---------------- per-feature sum / sumsq reduction ----------------
__global__ void k_reduce(const float* __restrict__ z, float* stats, int nrows) {
  __shared__ float ssum[256];
  __shared__ float ssq[256];
  int f  = threadIdx.x & 63;
  int rr = threadIdx.x >> 6;  // 0..3
  float s = 0.0f, q = 0.0f;
  for (int r = blockIdx.x * 4 + rr; r < nrows; r += gridDim.x * 4) {
    float v = z[(size_t)r * 64 + f];
    s += v;
    q += v * v;
  }
  ssum[threadIdx.x] = s;
  ssq[threadIdx.x]  = q;
  __syncthreads();
  if (threadIdx.x < 64) {
    s = ssum[threadIdx.x] + ssum[threadIdx.x + 64] + ssum[threadIdx.x + 128] + ssum[threadIdx.x + 192];
    q = ssq[threadIdx.x]  + ssq[threadIdx.x + 64]  + ssq[threadIdx.x + 128]  + ssq[threadIdx.x + 192];
    atomicAdd(&stats[threadIdx.x], s);
    atomicAdd(&stats[64 + threadIdx.x], q);
  }
}

// ---------------- BatchNorm (training stats) + ReLU ----------------
__global__ void k_bnrelu(const float* __restrict__ zin, float* __restrict__ zout,
                         const float* __restrict__ stats,
                         const float* __restrict__ gamma, const float* __restrict__ beta,
                         int nrows, float invN) {
  size_t total  = (size_t)nrows * 64;
  size_t stride = (size_t)gridDim.x * blockDim.x;
  for (size_t i = (size_t)blockIdx.x * blockDim.x + threadIdx.x; i < total; i += stride) {
    int f = (int)(i & 63);
    float mu  = stats[f] * invN;
    float var = stats[64 + f] * invN - mu * mu;
    float rs  = rsqrtf(var + BN_EPS);
    float v   = (zin[i] - mu) * rs * gamma[f] + beta[f];
    zout[i] = v > 0.0f ? v : 0.0f;
  }
}

// ---------------- JK head: logits += h @ fcW(64x10) + fcb ----------------
__global__ void k_logits(const float* __restrict__ h, const float* __restrict__ fcW,
                         const float* __restrict__ fcb, float* logits, int nrows) {
  __shared__ float Ws[64 * NCLS];
  __shared__ float bsh[NCLS];
  for (int i = threadIdx.x; i < 64 * NCLS; i += blockDim.x) Ws[i] = fcW[i];
  if (threadIdx.x < NCLS) bsh[threadIdx.x] = fcb[threadIdx.x];
  __syncthreads();
  for (int n = blockIdx.x * blockDim.x + threadIdx.x; n < nrows; n += gridDim.x * blockDim.x) {
    float acc[NCLS];
#pragma unroll
    for (int c = 0; c < NCLS; ++c) acc[c] = bsh[c];
    const float* hp = h + (size_t)n * 64;
#pragma unroll
    for (int f = 0; f < 64; ++f) {
      float v = hp[f];
#pragma unroll
      for (int c = 0; c < NCLS; ++c) acc[c] += v * Ws[f * NCLS + c];
    }
    float* lp = logits + (size_t)n * NCLS;
#pragma unroll
    for (int c = 0; c < NCLS; ++c) lp[c] += acc[c];
  }
}

// ---------------- log_softmax over C=10 ----------------
__global__ void k_lsm(const float* __restrict__ logits, float* __restrict__ out, int nrows) {
  for (int n = blockIdx.x * blockDim.x + threadIdx.x; n < nrows; n += gridDim.x * blockDim.x) {
    const float* lp = logits + (size_t)n * NCLS;
    float mx = lp[0];
#pragma unroll
    for (int c = 1; c < NCLS; ++c) mx = fmaxf(mx, lp[c]);
    float s = 0.0f;
#pragma unroll
    for (int c = 0; c < NCLS; ++c) s += __expf(lp[c] - mx);
    float lse = mx + __logf(s);
#pragma unroll
    for (int c = 0; c < NCLS; ++c) out[(size_t)n * NCLS + c] = lp[c] - lse;
  }
}

extern "C" void kernel_launch(void* const* d_in, const int* in_sizes, int n_in,
                              void* d_out, int out_size, void* d_ws, size_t ws_size,
                              hipStream_t stream) {
  const float*     x   = (const float*)d_in[0];
  const long long* ei  = (const long long*)d_in[1];
  const float*     W1  = (const float*)d_in[2];
  const float*     b1  = (const float*)d_in[3];
  const float*     g1  = (const float*)d_in[4];
  const float*     be1 = (const float*)d_in[5];
  const float*     W2  = (const float*)d_in[6];
  const float*     b2  = (const float*)d_in[7];
  const float*     gbn = (const float*)d_in[8];
  const float*     bbn = (const float*)d_in[9];
  const float*     fcW = (const float*)d_in[10];
  const float*     fcb = (const float*)d_in[11];

  const int N = in_sizes[0] / 64;
  const int E = in_sizes[1] / 2;
  const int L = 3;
  const float invN = 1.0f / (float)N;

  float* h      = (float*)d_ws;
  float* zz     = h + (size_t)N * 64;
  float* logits = zz + (size_t)N * 64;
  float* stats  = logits + (size_t)N * NCLS;

  // h <- x
  hipMemcpyAsync(h, x, (size_t)N * 64 * sizeof(float), hipMemcpyDeviceToDevice, stream);

  // logits <- 0, then layer-0 head (uses x == h)
  k_zero<<<2048, 256, 0, stream>>>(logits, (long long)N * NCLS);
  k_logits<<<512, 256, 0, stream>>>(h, fcW, fcb, logits, N);

  const int strips = (N + 15) / 16;
  const int gemm_blocks = (strips + GEMM_WPB - 1) / GEMM_WPB;

  for (int i = 0; i < L; ++i) {
    // agg <- 0 ; agg[dst] += h[src]
    k_zero<<<4096, 256, 0, stream>>>(zz, (long long)N * 64);
    k_agg<<<4096, 256, 0, stream>>>(h, ei, zz, E);

    // z1 = (h + agg) @ W1[i] + b1[i]   (written in place into zz)
    k_gemm64<true><<<gemm_blocks, 32 * GEMM_WPB, 0, stream>>>(h, zz, W1 + (size_t)i * 64 * 64,
                                                              b1 + i * 64, zz, N);
    // inner BN + ReLU
    k_zero<<<1, 128, 0, stream>>>(stats, 128);
    k_reduce<<<1024, 256, 0, stream>>>(zz, stats, N);
    k_bnrelu<<<4096, 256, 0, stream>>>(zz, zz, stats, g1 + i * 64, be1 + i * 64, N, invN);

    // z2 = z1 @ W2[i] + b2[i]   (in place)
    k_gemm64<false><<<gemm_blocks, 32 * GEMM_WPB, 0, stream>>>(zz, (const float*)0,
                                                               W2 + (size_t)i * 64 * 64,
                                                               b2 + i * 64, zz, N);
    // outer BN + ReLU -> h (next layer features)
    k_zero<<<1, 128, 0, stream>>>(stats, 128);
    k_reduce<<<1024, 256, 0, stream>>>(zz, stats, N);
    k_bnrelu<<<4096, 256, 0, stream>>>(zz, h, stats, gbn + i * 64, bbn + i * 64, N, invN);

    // JK head for this layer's output
    k_logits<<<512, 256, 0, stream>>>(h, fcW + (size_t)(i + 1) * 64 * NCLS,
                                      fcb + (i + 1) * NCLS, logits, N);
  }

  k_lsm<<<(N + 255) / 256, 256, 0, stream>>>(logits, (float*)d_out, N);
}